// Basic_Operator_59365037965641
// MI455X (gfx1250) — compile-verified
//
#include <hip/hip_runtime.h>

// Elementwise fusion:
//   out = w0*(x+y) + w1*(x*y) + w2*(x/(|y|+eps)) + w3*(y/(|x|+eps))
//       + w4*(x*sin(y)) + w5*(y*sin(x))
// with w[i] = sum_j softmax(param, axis=0)[i, j],  param: [6,16] row-major.
//
// Memory-bound streaming kernel: 12 B/elem over 67.1M elems => ~805 MB,
// ~34.5 us floor at 23.3 TB/s HBM. All big-tensor traffic uses 128-bit
// non-temporal global ops (no reuse; working set > 192 MB L2).

typedef float v4f __attribute__((ext_vector_type(4)));

#define EPS_F 1e-8f
#define TPB 256   // 8 wave32 per block
#define VPT 4     // float4 per thread -> 1024 float4 (16 KB per buffer) per block

// ---- tiny pre-kernel: collapse param[6,16] into 6 scalar weights in d_ws ----
__global__ void bo_weights_kernel(const float* __restrict__ param,
                                  float* __restrict__ w) {
    if (threadIdx.x != 0 || blockIdx.x != 0) return;
    float acc[6] = {0.f, 0.f, 0.f, 0.f, 0.f, 0.f};
    for (int j = 0; j < 16; ++j) {          // softmax over ops axis, per column
        float m = param[j];
        for (int i = 1; i < 6; ++i) m = fmaxf(m, param[i * 16 + j]);
        float e[6];
        float s = 0.f;
        for (int i = 0; i < 6; ++i) {
            e[i] = expf(param[i * 16 + j] - m);   // precise expf: runs once, off critical path
            s += e[i];
        }
        float inv = 1.0f / s;
        for (int i = 0; i < 6; ++i) acc[i] += e[i] * inv;
    }
    for (int i = 0; i < 6; ++i) w[i] = acc[i];
}

// ---- main streaming kernel ----
__global__ __launch_bounds__(TPB) void bo_fused_kernel(const v4f* __restrict__ x,
                                                       const v4f* __restrict__ y,
                                                       const float* __restrict__ w,
                                                       v4f* __restrict__ out,
                                                       int n4) {
    // Uniform address -> scalar (s_load) reads of the 6 weights.
    const float w0 = w[0], w1 = w[1], w2 = w[2];
    const float w3 = w[3], w4 = w[4], w5 = w[5];

    const int base = blockIdx.x * (TPB * VPT) + threadIdx.x;

    v4f xv[VPT], yv[VPT];
    bool ok[VPT];

    // Issue all 8 x/y 128-bit NT loads up front for memory-level parallelism.
#pragma unroll
    for (int k = 0; k < VPT; ++k) {
        const int idx = base + k * TPB;
        ok[k] = (idx < n4);
        if (ok[k]) {
            xv[k] = __builtin_nontemporal_load(&x[idx]);
            yv[k] = __builtin_nontemporal_load(&y[idx]);
        }
    }

#pragma unroll
    for (int k = 0; k < VPT; ++k) {
        if (!ok[k]) continue;
        v4f r;
#pragma unroll
        for (int c = 0; c < 4; ++c) {
            const float xs = xv[k][c];
            const float ys = yv[k][c];
            const float ax = __builtin_fabsf(xs);
            const float ay = __builtin_fabsf(ys);
            // v_rcp_f32 (~1 ulp) + mul instead of IEEE divide sequence;
            // denominators are >= EPS so this is well conditioned.
            const float d1 = xs * __builtin_amdgcn_rcpf(ay + EPS_F);
            const float d2 = ys * __builtin_amdgcn_rcpf(ax + EPS_F);
            // native v_sin_f32 path (inputs are ~N(0,1): tiny range, high accuracy)
            const float sy = __sinf(ys);
            const float sx = __sinf(xs);

            float acc = w0 * (xs + ys);
            acc = fmaf(w1, xs * ys, acc);
            acc = fmaf(w2, d1, acc);
            acc = fmaf(w3, d2, acc);
            acc = fmaf(w4, xs * sy, acc);
            acc = fmaf(w5, ys * sx, acc);
            r[c] = acc;
        }
        __builtin_nontemporal_store(r, &out[base + k * TPB]);
    }
}

extern "C" void kernel_launch(void* const* d_in, const int* in_sizes, int n_in,
                              void* d_out, int out_size, void* d_ws, size_t ws_size,
                              hipStream_t stream) {
    const float* x = (const float*)d_in[0];      // [4,4096,4096] f32
    const float* y = (const float*)d_in[1];      // [4,4096,4096] f32
    const float* param = (const float*)d_in[2];  // [6,16] f32
    float* out = (float*)d_out;
    float* w = (float*)d_ws;                     // 6 floats of scratch

    const int n = in_sizes[0];                   // 67,108,864 (divisible by 4)
    const int n4 = n / 4;

    bo_weights_kernel<<<1, 32, 0, stream>>>(param, w);

    const int per_block = TPB * VPT;             // 1024 float4 per block
    const int blocks = (n4 + per_block - 1) / per_block;
    bo_fused_kernel<<<blocks, TPB, 0, stream>>>((const v4f*)x, (const v4f*)y, w,
                                                (v4f*)out, n4);
}